// LineGraphTGCN_74380243632619
// MI455X (gfx1250) — compile-verified
//
#include <hip/hip_runtime.h>
#include <math.h>

#define NNODES 100000
#define NEDGES 1600000
#define FDIM   64
#define HID    32
#define NCLS   10
#define GCOLS  96   // 3*HID concatenated gate outputs z|r|h

typedef __attribute__((ext_vector_type(2))) float v2f;
typedef __attribute__((ext_vector_type(8))) float v8f;

__device__ __forceinline__ v8f wmma_f32_16x16x4(v2f a, v2f b, v8f c) {
  // 8 args: (neg_a, A, neg_b, B, c_mod, C, reuse_a, reuse_b)
  return __builtin_amdgcn_wmma_f32_16x16x4_f32(false, a, false, b, (short)0, c,
                                               false, false);
}

// ---------------------------------------------------------------- degree prep
__global__ __launch_bounds__(256) void k_zero(float* __restrict__ deg) {
  int i = blockIdx.x * 256 + threadIdx.x;
  if (i < NNODES) deg[i] = 0.0f;
}

__global__ __launch_bounds__(256) void k_degacc(const int* __restrict__ ecol,
                                                const float* __restrict__ ew,
                                                float* __restrict__ deg) {
  int e = blockIdx.x * 256 + threadIdx.x;
  if (e < NEDGES) atomicAdd(&deg[ecol[e]], ew[e]);
}

__global__ __launch_bounds__(256) void k_dis(float* __restrict__ deg) {
  int i = blockIdx.x * 256 + threadIdx.x;
  if (i < NNODES) deg[i] = rsqrtf(deg[i] + 1.0f);   // in place: deg -> dis
}

// ------------------------------------------- xw = x @ [Wz|Wr|Wh], fused agg init
// One wave per 16x16 output tile. agg = dis^2 * xw + bias (self-loop term).
__global__ __launch_bounds__(256) void k_xw(
    const float* __restrict__ x, const float* __restrict__ Wz,
    const float* __restrict__ Wr, const float* __restrict__ Wh,
    const float* __restrict__ bz, const float* __restrict__ br,
    const float* __restrict__ bh, const float* __restrict__ dis,
    float* __restrict__ xw, float* __restrict__ agg) {
  const int wid = threadIdx.x >> 5, lane = threadIdx.x & 31;
  const int tile = blockIdx.x * 8 + wid;
  const int COLT = GCOLS / 16;                    // 6
  if (tile >= (NNODES / 16) * COLT) return;       // wave-uniform exit
  const int rt = tile / COLT, ct = tile % COLT;
  const int m0 = rt * 16, n0 = ct * 16;
  const int half = lane >> 4, l16 = lane & 15;
  const int col = n0 + l16;
  const int gate = col >> 5, wcol = col & 31;
  const float* W = (gate == 0) ? Wz : (gate == 1) ? Wr : Wh;
  const float* bias = (gate == 0) ? bz : (gate == 1) ? br : bh;

  const float* arow = x + (m0 + l16) * FDIM + 2 * half;
  v8f c = {};
#pragma unroll
  for (int ks = 0; ks < 16; ++ks) {
    v2f a = *(const v2f*)(arow + 4 * ks);         // A[l16, 4ks+2*half+{0,1}]
    const int kr = 4 * ks + 2 * half;
    v2f b;
    b.x = W[kr * HID + wcol];                     // B[kr,   col]
    b.y = W[(kr + 1) * HID + wcol];               // B[kr+1, col]
    c = wmma_f32_16x16x4(a, b, c);
  }
  const float bv = bias[wcol];
#pragma unroll
  for (int v = 0; v < 8; ++v) {
    const int row = m0 + v + 8 * half;
    const float d = dis[row];
    const float val = c[v];
    xw[row * GCOLS + col] = val;
    agg[row * GCOLS + col] = d * d * val + bv;
  }
}

// ------------------------------------------------------- edge gather/scatter
// One wave per edge; lanes cover 96 columns in 3 strided atomic adds.
__global__ __launch_bounds__(256) void k_edge(
    const int* __restrict__ erow, const int* __restrict__ ecol,
    const float* __restrict__ ew, const float* __restrict__ dis,
    const float* __restrict__ xw, float* __restrict__ agg) {
  const int wid = threadIdx.x >> 5, lane = threadIdx.x & 31;
  const int e = blockIdx.x * 8 + wid;
  if (e >= NEDGES) return;
  const int r = erow[e], cc = ecol[e];
  const float nrm = dis[r] * ew[e] * dis[cc];
  const float* src = xw + (size_t)r * GCOLS;
  float* dst = agg + (size_t)cc * GCOLS;
#pragma unroll
  for (int t = 0; t < 3; ++t) {
    const int j = lane + 32 * t;
    atomicAdd(dst + j, nrm * src[j]);
  }
}

// ------------------------------- Z,R = sigmoid(concat(gcn_gate, h) @ LW + Lb)
__global__ __launch_bounds__(256) void k_zr(
    const float* __restrict__ agg, const float* __restrict__ h,
    const float* __restrict__ LzW, const float* __restrict__ Lzb,
    const float* __restrict__ LrW, const float* __restrict__ Lrb,
    float* __restrict__ Z, float* __restrict__ R) {
  const int wid = threadIdx.x >> 5, lane = threadIdx.x & 31;
  const int tile = blockIdx.x * 8 + wid;          // (N/16) * 4 tiles
  if (tile >= (NNODES / 16) * 4) return;
  const int rt = tile >> 2, combo = tile & 3;
  const int g = combo >> 1, ct = combo & 1;
  const int m0 = rt * 16, n0 = ct * 16;
  const int half = lane >> 4, l16 = lane & 15;
  const float* LW = g ? LrW : LzW;
  const float* Lb = g ? Lrb : Lzb;
  float* out = g ? R : Z;
  const int arow = m0 + l16;
  const float* ap0 = agg + (size_t)arow * GCOLS + g * HID + 2 * half; // k<32
  const float* ap1 = h + (size_t)arow * HID + 2 * half;               // k>=32
  v8f c = {};
#pragma unroll
  for (int ks = 0; ks < 16; ++ks) {
    v2f a = (ks < 8) ? *(const v2f*)(ap0 + 4 * ks)
                     : *(const v2f*)(ap1 + 4 * (ks - 8));
    const int kr = 4 * ks + 2 * half;
    v2f b;
    b.x = LW[kr * HID + n0 + l16];
    b.y = LW[(kr + 1) * HID + n0 + l16];
    c = wmma_f32_16x16x4(a, b, c);
  }
  const float bv = Lb[n0 + l16];
#pragma unroll
  for (int v = 0; v < 8; ++v) {
    const int row = m0 + v + 8 * half;
    const float s = 1.0f / (1.0f + __expf(-(c[v] + bv)));
    out[(size_t)row * HID + n0 + l16] = s;
  }
}

// ------- H_tilde = tanh(concat(gcn_h, h*R) @ LhW + Lhb); fuse GRU + Cauchy act
__global__ __launch_bounds__(256) void k_ht(
    const float* __restrict__ agg, const float* __restrict__ h,
    const float* __restrict__ R, const float* __restrict__ Z,
    const float* __restrict__ LhW, const float* __restrict__ Lhb,
    float* __restrict__ hact) {
  const int wid = threadIdx.x >> 5, lane = threadIdx.x & 31;
  const int tile = blockIdx.x * 8 + wid;          // (N/16) * 2 tiles
  if (tile >= (NNODES / 16) * 2) return;
  const int rt = tile >> 1, ct = tile & 1;
  const int m0 = rt * 16, n0 = ct * 16;
  const int half = lane >> 4, l16 = lane & 15;
  const int arow = m0 + l16;
  const float* ap0 = agg + (size_t)arow * GCOLS + 2 * HID + 2 * half;
  const float* hrow = h + (size_t)arow * HID + 2 * half;
  const float* rrow = R + (size_t)arow * HID + 2 * half;
  v8f c = {};
#pragma unroll
  for (int ks = 0; ks < 16; ++ks) {
    v2f a;
    if (ks < 8) {
      a = *(const v2f*)(ap0 + 4 * ks);
    } else {
      v2f hv = *(const v2f*)(hrow + 4 * (ks - 8));
      v2f rv = *(const v2f*)(rrow + 4 * (ks - 8));
      a = hv * rv;                                // reset-gated hidden state
    }
    const int kr = 4 * ks + 2 * half;
    v2f b;
    b.x = LhW[kr * HID + n0 + l16];
    b.y = LhW[(kr + 1) * HID + n0 + l16];
    c = wmma_f32_16x16x4(a, b, c);
  }
  const float bv = Lhb[n0 + l16];
#pragma unroll
  for (int v = 0; v < 8; ++v) {
    const int row = m0 + v + 8 * half;
    const int idx = row * HID + n0 + l16;
    const float ht = tanhf(c[v] + bv);
    const float z = Z[idx];
    const float hn = z * h[idx] + (1.0f - z) * ht;
    hact[idx] = hn / (hn * hn + 1.0f);            // Cauchy activation
  }
}

// ---------------------------------------------- y = h_act @ lin_W + lin_b
__global__ __launch_bounds__(256) void k_head(
    const float* __restrict__ hact, const float* __restrict__ linW,
    const float* __restrict__ linb, float* __restrict__ y) {
  const int wid = threadIdx.x >> 5, lane = threadIdx.x & 31;
  const int rt = blockIdx.x * 8 + wid;
  if (rt >= NNODES / 16) return;
  const int m0 = rt * 16;
  const int half = lane >> 4, l16 = lane & 15;
  const float* arow = hact + (size_t)(m0 + l16) * HID + 2 * half;
  v8f c = {};
#pragma unroll
  for (int ks = 0; ks < 8; ++ks) {                // K = 32
    v2f a = *(const v2f*)(arow + 4 * ks);
    const int kr = 4 * ks + 2 * half;
    v2f b;
    b.x = (l16 < NCLS) ? linW[kr * NCLS + l16] : 0.0f;       // zero-pad cols
    b.y = (l16 < NCLS) ? linW[(kr + 1) * NCLS + l16] : 0.0f;
    c = wmma_f32_16x16x4(a, b, c);
  }
  if (l16 < NCLS) {
    const float bv = linb[l16];
#pragma unroll
    for (int v = 0; v < 8; ++v) {
      const int row = m0 + v + 8 * half;
      y[row * NCLS + l16] = c[v] + bv;
    }
  }
}

extern "C" void kernel_launch(void* const* d_in, const int* in_sizes, int n_in,
                              void* d_out, int out_size, void* d_ws,
                              size_t ws_size, hipStream_t stream) {
  const float* x    = (const float*)d_in[0];
  const int*   ei   = (const int*)d_in[1];
  const float* ea   = (const float*)d_in[2];
  const float* h    = (const float*)d_in[3];
  const float* Wz   = (const float*)d_in[4];
  const float* bz   = (const float*)d_in[5];
  const float* Wr   = (const float*)d_in[6];
  const float* br   = (const float*)d_in[7];
  const float* Wh   = (const float*)d_in[8];
  const float* bh   = (const float*)d_in[9];
  const float* LzW  = (const float*)d_in[10];
  const float* Lzb  = (const float*)d_in[11];
  const float* LrW  = (const float*)d_in[12];
  const float* Lrb  = (const float*)d_in[13];
  const float* LhW  = (const float*)d_in[14];
  const float* Lhb  = (const float*)d_in[15];
  const float* linW = (const float*)d_in[16];
  const float* linb = (const float*)d_in[17];

  float* y    = (float*)d_out;                          // [N, 10]
  float* hact = (float*)d_out + (size_t)NNODES * NCLS;  // [N, 32]

  // workspace layout: dis | xw | agg | Z | R  (N*257 floats ~ 103 MB)
  float* ws  = (float*)d_ws;
  float* deg = ws;                                // N  (deg -> dis in place)
  float* xw  = ws + NNODES;                       // N*96
  float* agg = xw + (size_t)NNODES * GCOLS;       // N*96
  float* Z   = agg + (size_t)NNODES * GCOLS;      // N*32
  float* R   = Z + (size_t)NNODES * HID;          // N*32

  const int* erow = ei;
  const int* ecol = ei + NEDGES;

  k_zero<<<(NNODES + 255) / 256, 256, 0, stream>>>(deg);
  k_degacc<<<(NEDGES + 255) / 256, 256, 0, stream>>>(ecol, ea, deg);
  k_dis<<<(NNODES + 255) / 256, 256, 0, stream>>>(deg);

  const int xw_tiles = (NNODES / 16) * (GCOLS / 16);    // 37500
  k_xw<<<(xw_tiles + 7) / 8, 256, 0, stream>>>(x, Wz, Wr, Wh, bz, br, bh, deg,
                                               xw, agg);
  k_edge<<<NEDGES / 8, 256, 0, stream>>>(erow, ecol, ea, deg, xw, agg);

  const int zr_tiles = (NNODES / 16) * 4;               // 25000
  k_zr<<<(zr_tiles + 7) / 8, 256, 0, stream>>>(agg, h, LzW, Lzb, LrW, Lrb, Z, R);

  const int ht_tiles = (NNODES / 16) * 2;               // 12500
  k_ht<<<(ht_tiles + 7) / 8, 256, 0, stream>>>(agg, h, R, Z, LhW, Lhb, hact);

  k_head<<<(NNODES / 16 + 7) / 8, 256, 0, stream>>>(hact, linW, linb, y);
}